// SpatialAttention_21930103013486
// MI455X (gfx1250) — compile-verified
//
#include <hip/hip_runtime.h>

#define T_STEPS 24
#define BATCH   64
#define NI      32
#define IL      24
#define VEC     24
#define NS      256
#define HID     256
#define DIN     (NI + NS)   // 288
#define LAMDA   0.2f

typedef __attribute__((ext_vector_type(2))) float v2f;
typedef __attribute__((ext_vector_type(8))) float v8f;

__device__ __forceinline__ float fast_tanh(float x) {
    x = fminf(fmaxf(x, -15.f), 15.f);
    float e = __expf(2.f * x);
    return __fdividef(e - 1.f, e + 1.f);
}
__device__ __forceinline__ float sigm(float x) {
    return __fdividef(1.f, 1.f + __expf(-x));
}

__global__ __launch_bounds__(256, 2)
void spatial_attn_kernel(
    const float* __restrict__ local_inputs,   // [T,B,NI]
    const float* __restrict__ global_inputs,  // [T,B,NS]
    const float* __restrict__ local_states,   // [B,NI,IL] viewed [B,IL,NI]
    const float* __restrict__ global_states,  // [B,NS,NI,IL] viewed [B,IL,NS,NI]
    const float* __restrict__ distmat,        // [B,NS]
    const float* __restrict__ la0,            // [B,NI]
    const float* __restrict__ ga0,            // [B,NS]
    const float* __restrict__ Wcl,            // [VEC,IL]
    const float* __restrict__ bcl,            // [VEC]
    const float* __restrict__ Wll,            // [VEC,HID]
    const float* __restrict__ bll,            // [VEC]
    const float* __restrict__ vl,             // [VEC]
    const float* __restrict__ Wcg,            // [VEC,IL,NI] = [24,768]
    const float* __restrict__ bcg,            // [VEC]
    const float* __restrict__ Wlg,            // [VEC,HID]
    const float* __restrict__ blg,            // [VEC]
    const float* __restrict__ vg,             // [VEC]
    const float* __restrict__ Wih,            // [4H, DIN]
    const float* __restrict__ bih,            // [4H]
    const float* __restrict__ bhh,            // [4H]
    float* __restrict__ outs,                 // [T,B,H]
    float* __restrict__ cs_out,               // [B,H]
    float* __restrict__ las,                  // [T,B,NI]
    float* __restrict__ gas)                  // [T,B,NS]
{
    const int b    = blockIdx.x;
    const int tid  = threadIdx.x;
    const int lane = tid & 31;
    const int wv   = tid >> 5;

    __shared__ float hf_g[VEC][NS];          // 24 KB
    __shared__ float hf_l[VEC][NI];          // 3 KB
    __shared__ __align__(16) float xbuf[DIN];
    __shared__ float cbuf[HID];
    __shared__ float ybuf[48];               // y_l[0:24], y_g[24:48]
    __shared__ float vbuf[48];               // vl, vg
    __shared__ float laS[NI];
    __shared__ float gaS[NS];
    __shared__ float red[256];
    __shared__ float slbuf[NI];
    __shared__ float scal[2];

    // ---------------- init carry & small vectors ----------------
    if (tid < NI) laS[tid] = la0[b * NI + tid];
    gaS[tid] = ga0[b * NS + tid];
    if (tid < 24)       vbuf[tid] = vl[tid];
    else if (tid < 48)  vbuf[tid] = vg[tid - 24];

    // ---------------- hf_l[o][l] = bcl[o] + sum_c Wcl[o,c]*lhid[c,l] ----------------
    {
        const float* lsb = local_states + b * (NI * IL);   // flat [768], lhid[c,l] = lsb[c*32+l]
        for (int idx = tid; idx < VEC * NI; idx += 256) {
            int o = idx >> 5, l = idx & 31;
            float acc = bcl[o];
            #pragma unroll
            for (int c = 0; c < IL; ++c)
                acc = fmaf(Wcl[o * IL + c], lsb[c * NI + l], acc);
            hf_l[o][l] = acc;
        }
    }

    // ---------------- hf_g via V_WMMA_F32_16X16X4_F32 ----------------
    // C[o=24, s=256] = Wcg[24,768] x Bm[768,256], Bm[k=c*32+j, s] = gs[b, c, s, j]
    {
        const float* gsb   = global_states + (size_t)b * (NS * NI * IL);
        const int    hl    = lane & 15;
        const int    koff  = (lane < 16) ? 0 : 2;
        const float  amask = (hl < 8) ? 1.f : 0.f;          // rows 24..31 of tile-1 are padding
        const int    m0    = hl;                            // M tile 0: rows 0..15
        const int    m1    = (hl < 8) ? (16 + hl) : 23;     // M tile 1: rows 16..23 (clamped)

        for (int nt = wv; nt < 16; nt += 8) {               // each wave: 2 N-tiles of 16
            const int s = nt * 16 + hl;
            v8f acc0 = {};
            v8f acc1 = {};
            for (int k = 0; k < 768; k += 4) {
                const int kk = k + koff;
                const int ci = kk >> 5;
                const int j  = kk & 31;                     // even -> (j, j+1) contiguous
                v2f bv = *(const v2f*)(gsb + ci * (NS * NI) + s * NI + j);
                v2f a0 = *(const v2f*)(Wcg + m0 * 768 + kk);
                v2f a1 = *(const v2f*)(Wcg + m1 * 768 + kk);
                a1.x *= amask; a1.y *= amask;
                acc0 = __builtin_amdgcn_wmma_f32_16x16x4_f32(
                           false, a0, false, bv, (short)0, acc0, false, false);
                acc1 = __builtin_amdgcn_wmma_f32_16x16x4_f32(
                           false, a1, false, bv, (short)0, acc1, false, false);
            }
            // C/D layout: VGPR r -> lanes 0-15: M=r, lanes 16-31: M=r+8 ; N = s
            #pragma unroll
            for (int r = 0; r < 8; ++r) {
                const int o0 = (lane < 16) ? r : (r + 8);
                hf_g[o0][s] = acc0[r] + bcg[o0];
                if (lane < 16)                               // tile-1 valid rows: 16..23
                    hf_g[16 + r][s] = acc1[r] + bcg[16 + r];
            }
        }
    }

    // ---------------- hoisted per-thread scan constants ----------------
    // torch gate order i,f,g,o ; f is dead (c0 == 0)
    const float bi = bih[tid]           + bhh[tid];
    const float bg = bih[tid + 2 * HID] + bhh[tid + 2 * HID];
    const float bo = bih[tid + 3 * HID] + bhh[tid + 3 * HID];
    const float dist_s = distmat[b * NS + tid];
    const float* Wi_i = Wih + (size_t)(tid)           * DIN;
    const float* Wi_g = Wih + (size_t)(tid + 2 * HID) * DIN;
    const float* Wi_o = Wih + (size_t)(tid + 3 * HID) * DIN;
    __builtin_prefetch(Wi_i, 0, 1);
    __builtin_prefetch(Wi_g, 0, 1);
    __builtin_prefetch(Wi_o, 0, 1);

    __syncthreads();

    // ---------------- recurrent scan over T ----------------
    for (int t = 0; t < T_STEPS; ++t) {
        // x = [la * l_inp, ga * g_inp]
        if (tid < NI)
            xbuf[tid] = laS[tid] * local_inputs[((size_t)t * BATCH + b) * NI + tid];
        xbuf[NI + tid] = gaS[tid] * global_inputs[((size_t)t * BATCH + b) * NS + tid];
        __syncthreads();

        // gates: 3 live rows per thread, 288-length dots (Wih streamed from L2)
        float ai = bi, ag = bg, ao = bo;
        for (int k = 0; k < DIN; k += 4) {
            const float4 xk = *(const float4*)(&xbuf[k]);
            const float4 wi = *(const float4*)(Wi_i + k);
            const float4 wg = *(const float4*)(Wi_g + k);
            const float4 wo = *(const float4*)(Wi_o + k);
            ai = fmaf(wi.x, xk.x, ai); ai = fmaf(wi.y, xk.y, ai);
            ai = fmaf(wi.z, xk.z, ai); ai = fmaf(wi.w, xk.w, ai);
            ag = fmaf(wg.x, xk.x, ag); ag = fmaf(wg.y, xk.y, ag);
            ag = fmaf(wg.z, xk.z, ag); ag = fmaf(wg.w, xk.w, ag);
            ao = fmaf(wo.x, xk.x, ao); ao = fmaf(wo.y, xk.y, ao);
            ao = fmaf(wo.z, xk.z, ao); ao = fmaf(wo.w, xk.w, ao);
        }
        const float cv = sigm(ai) * fast_tanh(ag);
        const float hv = sigm(ao) * fast_tanh(cv);
        cbuf[tid] = cv;
        outs[((size_t)t * BATCH + b) * HID + tid] = hv;
        if (t == T_STEPS - 1) cs_out[b * HID + tid] = cv;
        __syncthreads();

        // y_l / y_g : 48 dots of length 256 (q = c)
        if (tid < 48) {
            const int    d  = (tid < 24) ? tid : (tid - 24);
            const float* Wr = (tid < 24) ? (Wll + d * HID) : (Wlg + d * HID);
            float acc = (tid < 24) ? bll[d] : blg[d];
            for (int n = 0; n < HID; ++n)
                acc = fmaf(Wr[n], cbuf[n], acc);
            ybuf[tid] = acc;
        }
        __syncthreads();

        // s_l[l] = sum_{o,d} tanh(hf_l[o,l] + y_l[d]) * vl[d] ; partials over o-groups
        {
            const int l = tid & 31, og = tid >> 5;        // og covers o = 3*og .. 3*og+2
            float acc = 0.f;
            #pragma unroll
            for (int oi = 0; oi < 3; ++oi) {
                const float hvv = hf_l[og * 3 + oi][l];
                #pragma unroll
                for (int d = 0; d < 24; ++d)
                    acc = fmaf(fast_tanh(hvv + ybuf[d]), vbuf[d], acc);
            }
            red[tid] = acc;
        }
        __syncthreads();
        if (tid < NI) {
            float s = 0.f;
            #pragma unroll
            for (int og = 0; og < 8; ++og) s += red[og * 32 + tid];
            slbuf[tid] = s;
        }
        __syncthreads();

        // s_g[s] per thread: 24x24 tanh over LDS hf_g (s-major -> conflict-free)
        float sg = 0.f;
        for (int o = 0; o < 24; ++o) {
            const float hvv = hf_g[o][tid];
            #pragma unroll
            for (int d = 0; d < 24; ++d)
                sg = fmaf(fast_tanh(hvv + ybuf[24 + d]), vbuf[24 + d], sg);
        }
        sg = (1.f - LAMDA) * sg + LAMDA * dist_s;

        // softmax over NI=32 (serial, deterministic)
        if (tid == 0) {
            float mx = slbuf[0];
            for (int i = 1; i < NI; ++i) mx = fmaxf(mx, slbuf[i]);
            float sum = 0.f;
            for (int i = 0; i < NI; ++i) sum += __expf(slbuf[i] - mx);
            scal[0] = mx; scal[1] = sum;
        }
        __syncthreads();
        if (tid < NI) {
            const float e = __fdividef(__expf(slbuf[tid] - scal[0]), scal[1]);
            laS[tid] = e;
            las[((size_t)t * BATCH + b) * NI + tid] = e;
        }

        // softmax over NS=256 (fixed-order LDS trees)
        red[tid] = sg;
        __syncthreads();
        for (int st = 128; st >= 1; st >>= 1) {
            if (tid < st) red[tid] = fmaxf(red[tid], red[tid + st]);
            __syncthreads();
        }
        const float mxg = red[0];
        __syncthreads();
        const float eg = __expf(sg - mxg);
        red[tid] = eg;
        __syncthreads();
        for (int st = 128; st >= 1; st >>= 1) {
            if (tid < st) red[tid] += red[tid + st];
            __syncthreads();
        }
        const float gv = __fdividef(eg, red[0]);
        gaS[tid] = gv;
        gas[((size_t)t * BATCH + b) * NS + tid] = gv;
        __syncthreads();
    }
}

extern "C" void kernel_launch(void* const* d_in, const int* in_sizes, int n_in,
                              void* d_out, int out_size, void* d_ws, size_t ws_size,
                              hipStream_t stream) {
    const float* local_inputs  = (const float*)d_in[0];
    const float* global_inputs = (const float*)d_in[1];
    const float* local_states  = (const float*)d_in[2];
    const float* global_states = (const float*)d_in[3];
    const float* distmat       = (const float*)d_in[4];
    const float* la0           = (const float*)d_in[5];
    const float* ga0           = (const float*)d_in[6];
    const float* Wcl           = (const float*)d_in[7];
    const float* bcl           = (const float*)d_in[8];
    const float* Wll           = (const float*)d_in[9];
    const float* bll           = (const float*)d_in[10];
    const float* vl            = (const float*)d_in[11];
    const float* Wcg           = (const float*)d_in[12];
    const float* bcg           = (const float*)d_in[13];
    const float* Wlg           = (const float*)d_in[14];
    const float* blg           = (const float*)d_in[15];
    const float* vg            = (const float*)d_in[16];
    const float* Wih           = (const float*)d_in[17];
    const float* bih           = (const float*)d_in[18];
    const float* bhh           = (const float*)d_in[19];

    float* out  = (float*)d_out;
    float* outs = out;                                           // T*B*H
    float* cs   = outs + (size_t)T_STEPS * BATCH * HID;          // B*H
    float* las  = cs   + (size_t)BATCH * HID;                    // T*B*NI
    float* gas  = las  + (size_t)T_STEPS * BATCH * NI;           // T*B*NS

    spatial_attn_kernel<<<dim3(BATCH), dim3(256), 0, stream>>>(
        local_inputs, global_inputs, local_states, global_states, distmat,
        la0, ga0, Wcl, bcl, Wll, bll, vl, Wcg, bcg, Wlg, blg, vg,
        Wih, bih, bhh, outs, cs, las, gas);
}